// DenseEdgeEncoder_49237505081544
// MI455X (gfx1250) — compile-verified
//
#include <hip/hip_runtime.h>
#include <cstdint>

// Problem constants (match the reference module-level constants).
#define NNODE 256   // MAX_NODES
#define DIM   64    // EMB_DIM

// True clang vector types (HIP's float4 class is rejected by the builtins).
typedef float fvec4 __attribute__((ext_vector_type(4)));
typedef int   v2i   __attribute__((ext_vector_type(2)));
typedef __attribute__((address_space(1))) v2i gv2i_t;  // global 64-bit chunk
typedef __attribute__((address_space(3))) v2i lv2i_t;  // LDS 64-bit chunk

#if defined(__AMDGCN__) && defined(__gfx1250__)
#  if __has_builtin(__builtin_amdgcn_global_store_async_from_lds_b64)
#    define USE_ASYNC_FILL 1
#  endif
#endif

// ---------------------------------------------------------------------------
// Kernel 1: fill the whole [B,N,N,D] output.
//   off-diagonal, non-edge -> emb_table[2]   (background)
//   diagonal               -> emb_table[1]
// Edge rows are overwritten later by scatter_edges.
//
// gfx1250 path: stage both template rows (256 B each) in LDS, then issue one
// GLOBAL_STORE_ASYNC_FROM_LDS_B64 per output row: 32 lanes x 8 B = 256 B row
// DMA'd from LDS, tracked on ASYNCcnt. One instruction retires a full row.
// ---------------------------------------------------------------------------
__global__ __launch_bounds__(256) void fill_bg(const float* __restrict__ emb,
                                               float* __restrict__ out,
                                               int rows) {
  __shared__ __align__(16) float lds[128];  // [0:64) = row2 (bg), [64:128) = row1 (diag)
  const int tid = threadIdx.x;
  if (tid < 64)        lds[tid] = emb[2 * DIM + tid];
  else if (tid < 128)  lds[tid] = emb[1 * DIM + (tid - 64)];
  __syncthreads();

#if defined(USE_ASYNC_FILL)
  const int lane = tid & 31;
  const int wave = (int)((blockIdx.x * blockDim.x + (unsigned)tid) >> 5);
  const int nw   = (int)((gridDim.x * blockDim.x) >> 5);
  const unsigned int base = (unsigned int)(uintptr_t)&lds[0];
  const unsigned int lds_bg   = base + (unsigned)lane * 8u;   // bytes [8*lane, 8*lane+8) of row2
  const unsigned int lds_diag = lds_bg + 256u;                // same bytes of row1
  for (int row = wave; row < rows; row += nw) {
    const int j = row & (NNODE - 1);
    const int i = (row >> 8) & (NNODE - 1);
    const unsigned int lsrc = (i == j) ? lds_diag : lds_bg;
    float* gp = out + (size_t)row * DIM + lane * 2;  // 8 bytes per lane
    __builtin_amdgcn_global_store_async_from_lds_b64(
        (gv2i_t*)(uintptr_t)gp, (lv2i_t*)lsrc, /*offset=*/0, /*cpol=*/0);
  }
#  if __has_builtin(__builtin_amdgcn_s_wait_asynccnt)
  __builtin_amdgcn_s_wait_asynccnt(0);
#  else
  asm volatile("s_wait_asynccnt 0" ::: "memory");
#  endif
#else
  // Fallback: vectorized non-temporal b128 stores.
  const size_t gtid   = blockIdx.x * (size_t)blockDim.x + tid;
  const size_t stride = (size_t)gridDim.x * blockDim.x;  // multiple of 16 -> chunk id fixed
  const int c = (int)(gtid & 15);                        // which float4 of the row
  const fvec4* lv = (const fvec4*)lds;
  const fvec4 r_bg   = lv[c];
  const fvec4 r_diag = lv[16 + c];
  fvec4* o4 = (fvec4*)out;
  const size_t tot = (size_t)rows * 16;
  for (size_t idx = gtid; idx < tot; idx += stride) {
    const size_t row = idx >> 4;
    const int j = (int)(row & (NNODE - 1));
    const int i = (int)((row >> 8) & (NNODE - 1));
    __builtin_nontemporal_store((i == j) ? r_diag : r_bg, o4 + idx);
  }
#endif
}

// ---------------------------------------------------------------------------
// Kernel 2: overwrite edge rows with edge_attr.
// 16 threads per edge; each copies one 16-byte chunk of the 64-float row.
// Edges are distinct (src,dst) pairs and table[0]==0, so plain overwrite is
// exactly edge_attr + table[0].
// ---------------------------------------------------------------------------
__global__ __launch_bounds__(256) void scatter_edges(
    const float* __restrict__ edge_attr, const int* __restrict__ eidx,
    const int* __restrict__ batch, float* __restrict__ out, int etot) {
  const int t = blockIdx.x * blockDim.x + threadIdx.x;
  if (t >= etot * 16) return;
  const int e = t >> 4;
  const int c = t & 15;
  const int src = eidx[e];
  const int dst = eidx[etot + e];
  const int g  = batch[src];              // graph id (pyG batch vector)
  const int ls = src - g * NNODE;         // starts[g] == g*NNODE (equal-size graphs)
  const int ld = dst - g * NNODE;
  const size_t row = ((size_t)g * NNODE + (size_t)ls) * NNODE + (size_t)ld;
  const fvec4 v = __builtin_nontemporal_load((const fvec4*)edge_attr + (size_t)e * 16 + c);
  __builtin_nontemporal_store(v, (fvec4*)out + row * 16 + c);
}

extern "C" void kernel_launch(void* const* d_in, const int* in_sizes, int n_in,
                              void* d_out, int out_size, void* d_ws, size_t ws_size,
                              hipStream_t stream) {
  const float* edge_attr = (const float*)d_in[0];  // [Etot, 64] f32
  const float* emb       = (const float*)d_in[1];  // [3, 64]    f32
  const int*   eidx      = (const int*)d_in[2];    // [2, Etot]  i32
  const int*   batch     = (const int*)d_in[3];    // [B*N]      i32
  float*       out       = (float*)d_out;          // [B,N,N,64] f32

  const int etot = in_sizes[2] / 2;                // 65536
  const int rows = out_size / DIM;                 // 1048576 rows of 64 floats

  // Phase 1: structured fill (background + diagonal), memory-bandwidth bound.
  fill_bg<<<2048, 256, 0, stream>>>(emb, out, rows);

  // Phase 2: scatter edge feature rows (ordered after fill on the stream).
  const int nthreads = etot * 16;
  scatter_edges<<<(nthreads + 255) / 256, 256, 0, stream>>>(edge_attr, eidx, batch, out, etot);
}